// HCSPN_Model_72232759984839
// MI455X (gfx1250) — compile-verified
//
#include <hip/hip_runtime.h>

#define EPSF 1e-6f

// ---- CDNA5 async global->LDS path (probe-verified on this toolchain) ------
#if defined(__has_builtin)
#  if __has_builtin(__builtin_amdgcn_global_load_async_to_lds_b32) && \
      __has_builtin(__builtin_amdgcn_s_wait_asynccnt)
#    define USE_ASYNC_LDS 1
#  endif
#endif

#ifdef USE_ASYNC_LDS
// Signature (from round-1 diagnostic): (AS1 int* src, AS3 int* dst, imm, imm)
typedef __attribute__((address_space(1))) int GASI;
typedef __attribute__((address_space(3))) int LASI;
#endif

// ---- avg_pool2 applied twice == mean over 4x4 block -----------------------
template <int Ho, int Wo, int Hi, int Wi>
__global__ __launch_bounds__(256) void hcspn_down4(
    const float* __restrict__ in, float* __restrict__ out)
{
  const int N = 8 * Ho * Wo;
  int idx = blockIdx.x * 256 + threadIdx.x;
  if (idx >= N) return;
  int x = idx % Wo; int t = idx / Wo;          // const divisors -> mulhi/shift
  int y = t % Ho;   int b = t / Ho;
  const float* p = in + ((size_t)b * Hi + (size_t)y * 4) * Wi + (size_t)x * 4;
  float s = 0.f;
#pragma unroll
  for (int r = 0; r < 4; ++r)
#pragma unroll
    for (int c = 0; c < 4; ++c) s += p[(size_t)r * Wi + c];
  out[idx] = s * (1.0f / 16.0f);
}

// ---- nearest 2x upsample --------------------------------------------------
template <int Ho, int Wo>
__global__ __launch_bounds__(256) void hcspn_up2(
    const float* __restrict__ in, float* __restrict__ out)
{
  const int Hi = Ho / 2, Wi = Wo / 2;
  const int N = 8 * Ho * Wo;
  int idx = blockIdx.x * 256 + threadIdx.x;
  if (idx >= N) return;
  int x = idx % Wo; int t = idx / Wo;
  int y = t % Ho;   int b = t / Ho;
  out[idx] = in[((size_t)b * Hi + (y >> 1)) * Wi + (x >> 1)];
}

// ---- one fused propagation iteration --------------------------------------
// feat_in [B,1,H,W], guid [B,24,H,W], conf_j = conf base + j*H*W (batch
// stride 6*H*W), aff[18]. Block = 32x8 pixels, LDS holds 36x12 halo tile.
// NT: stream guidance non-temporally (scale 2: 236MB > 192MB L2, no reuse).
template <int H, int W, bool NT>
__global__ __launch_bounds__(256) void hcspn_prop(
    const float* __restrict__ feat_in,
    const float* __restrict__ guid,
    const float* __restrict__ conf_j,
    const float* __restrict__ aff,
    float* __restrict__ feat_out,
    int aff_idx)
{
  constexpr size_t HW = (size_t)H * W;
  __shared__ float sm[12][36];
  const int lx = threadIdx.x, ly = threadIdx.y;
  const int bx = blockIdx.x * 32, by = blockIdx.y * 8;
  const int b  = blockIdx.z;
  const float* fin = feat_in + (size_t)b * HW;

  // Stage zero-padded 36x12 feat tile into LDS (async DMA on CDNA5).
  const int t = ly * 32 + lx;
#pragma unroll
  for (int i = 0; i < 2; ++i) {
    int e = t + i * 256;
    if (e < 36 * 12) {
      int sy = e / 36, sx = e - sy * 36;
      int gy = by + sy - 2, gx = bx + sx - 2;
      bool ok = (gy >= 0) && (gy < H) && (gx >= 0) && (gx < W);
#ifdef USE_ASYNC_LDS
      if (ok) {
        __builtin_amdgcn_global_load_async_to_lds_b32(
            (GASI*)(fin + (size_t)gy * W + gx),
            (LASI*)&sm[sy][sx], 0, 0);
      } else {
        sm[sy][sx] = 0.0f;
      }
#else
      sm[sy][sx] = ok ? fin[(size_t)gy * W + gx] : 0.0f;
#endif
    }
  }
#ifdef USE_ASYNC_LDS
  __builtin_amdgcn_s_wait_asynccnt(0);
#endif
  __syncthreads();

  const int x = bx + lx, y = by + ly;
  const float center = sm[ly + 2][lx + 2];
  const float* gpix = guid + (size_t)b * 24 * HW + (size_t)y * W + x;
  const float w  = __builtin_nontemporal_load(
      conf_j + (size_t)b * 6 * HW + (size_t)y * W + x);   // read-once: NT
  const float wB = 1.0f - w + EPSF;            // (1 - weight + eps)
  const float cp = aff[aff_idx] + EPSF;        // (const + eps)

  float gsum = 0.f, gabs = 0.f, gdot = 0.f, esum = 0.f, edot = 0.f;
#pragma unroll
  for (int m = 0; m < 24; ++m) {
    const int k  = (m < 12) ? m : (m + 1);     // skip center slot 12
    const int dy = k / 5, dx = k - dy * 5;
    const float p = sm[ly + dy][lx + dx];
    const float* gp = gpix + (size_t)m * HW;
    const float g = NT ? __builtin_nontemporal_load(gp) : *gp;
    gsum += g;
    gabs += fabsf(g);
    gdot += p * g;
    const float d = fabsf(center - p) + EPSF;  // abs_diff + eps
    const float e = __expf(-cp * d);           // v_exp_f32 (TRANS, co-exec)
    esum += e;
    edot += p * e;
  }
  // normalize_guide: denom = sum|g| + 2eps ; center = 1 - (sum g + eps)/denom
  const float invg = __builtin_amdgcn_rcpf(gabs + 2.0f * EPSF);
  const float ceng = 1.0f - (gsum + EPSF) * invg;
  const float invi = __builtin_amdgcn_rcpf(esum + 2.0f * EPSF);  // exp>0: abs==sum
  const float ceni = 1.0f - (esum + EPSF) * invi;

  const float out = (w * invg) * gdot + (wB * invi) * edot
                  + center * (ceng * w + ceni * wB);
  feat_out[(size_t)b * HW + (size_t)y * W + x] = out;
}

// ---------------------------------------------------------------------------
extern "C" void kernel_launch(void* const* d_in, const int* in_sizes, int n_in,
                              void* d_out, int out_size, void* d_ws, size_t ws_size,
                              hipStream_t stream) {
  (void)in_sizes; (void)n_in; (void)out_size; (void)ws_size;
  const float* depth   = (const float*)d_in[0];
  const float* guid[3] = {(const float*)d_in[1], (const float*)d_in[2], (const float*)d_in[3]};
  const float* conf[3] = {(const float*)d_in[4], (const float*)d_in[5], (const float*)d_in[6]};
  const float* aff     = (const float*)d_in[7];
  float* outp = (float*)d_out;

  const int B = 8, Hf = 480, Wf = 640;
  const size_t cap = (size_t)B * Hf * Wf;          // 2.46M floats per buffer
  float* bufA = (float*)d_ws;
  float* bufB = bufA + cap;

  // coarse_depth -> 1/4 resolution
  {
    int N = B * 120 * 160;
    hcspn_down4<120, 160, 480, 640>
        <<<dim3((N + 255) / 256), dim3(256), 0, stream>>>(depth, bufA);
  }

  float* cur = bufA;
  float* nxt = bufB;
  const int Hs[3] = {120, 240, 480};
  const int Ws[3] = {160, 320, 640};

  for (int i = 0; i < 3; ++i) {
    const int H = Hs[i], W = Ws[i];
    if (i >= 1) {
      int N = B * H * W;
      float* other = (cur == bufA) ? bufB : bufA;
      if (i == 1)
        hcspn_up2<240, 320><<<dim3((N + 255) / 256), dim3(256), 0, stream>>>(cur, other);
      else
        hcspn_up2<480, 640><<<dim3((N + 255) / 256), dim3(256), 0, stream>>>(cur, other);
      cur = other;
      nxt = (cur == bufA) ? bufB : bufA;
    }
    const size_t HW = (size_t)H * W;
    dim3 grid(W / 32, H / 8, B), block(32, 8, 1);
    for (int j = 0; j < 6; ++j) {
      const bool last = (i == 2 && j == 5);
      float* dst = last ? outp : nxt;
      const float* cj = conf[i] + (size_t)j * HW;
      const int idx = i * 6 + j;
      if (i == 0)
        hcspn_prop<120, 160, false><<<grid, block, 0, stream>>>(cur, guid[i], cj, aff, dst, idx);
      else if (i == 1)
        hcspn_prop<240, 320, false><<<grid, block, 0, stream>>>(cur, guid[i], cj, aff, dst, idx);
      else
        hcspn_prop<480, 640, true><<<grid, block, 0, stream>>>(cur, guid[i], cj, aff, dst, idx);
      if (!last) { float* tmp = cur; cur = nxt; nxt = tmp; }
    }
  }
}